// gen_gauss_24713241822147
// MI455X (gfx1250) — compile-verified
//
#include <hip/hip_runtime.h>
#include <math.h>

typedef float v2f __attribute__((ext_vector_type(2)));
typedef float v4f __attribute__((ext_vector_type(4)));
typedef float v8f __attribute__((ext_vector_type(8)));

#define B_ 8
#define N_ 262144
#define M_ 8192
#define H_ 512
#define W_ 512
#define XMIN_ 0.0f
#define YMIN_ (-25.6f)
#define ZMIN_ (-3.0f)
#define ZMAX_ 3.0f
#define INV_V 10.0f            // 1/0.1
#define GAUSS_K (1.0f / (2.0f * 1.5f * 1.5f))

// ---------------- pose_0to1 = inv(pose1) @ pose0, one thread per batch ----------
__device__ static void inv4x4(const float* m, float* o) {
    float inv[16];
    inv[0]  =  m[5]*m[10]*m[15] - m[5]*m[11]*m[14] - m[9]*m[6]*m[15] + m[9]*m[7]*m[14] + m[13]*m[6]*m[11] - m[13]*m[7]*m[10];
    inv[4]  = -m[4]*m[10]*m[15] + m[4]*m[11]*m[14] + m[8]*m[6]*m[15] - m[8]*m[7]*m[14] - m[12]*m[6]*m[11] + m[12]*m[7]*m[10];
    inv[8]  =  m[4]*m[9]*m[15]  - m[4]*m[11]*m[13] - m[8]*m[5]*m[15] + m[8]*m[7]*m[13] + m[12]*m[5]*m[11] - m[12]*m[7]*m[9];
    inv[12] = -m[4]*m[9]*m[14]  + m[4]*m[10]*m[13] + m[8]*m[5]*m[14] - m[8]*m[6]*m[13] - m[12]*m[5]*m[10] + m[12]*m[6]*m[9];
    inv[1]  = -m[1]*m[10]*m[15] + m[1]*m[11]*m[14] + m[9]*m[2]*m[15] - m[9]*m[3]*m[14] - m[13]*m[2]*m[11] + m[13]*m[3]*m[10];
    inv[5]  =  m[0]*m[10]*m[15] - m[0]*m[11]*m[14] - m[8]*m[2]*m[15] + m[8]*m[3]*m[14] + m[12]*m[2]*m[11] - m[12]*m[3]*m[10];
    inv[9]  = -m[0]*m[9]*m[15]  + m[0]*m[11]*m[13] + m[8]*m[1]*m[15] - m[8]*m[3]*m[13] - m[12]*m[1]*m[11] + m[12]*m[3]*m[9];
    inv[13] =  m[0]*m[9]*m[14]  - m[0]*m[10]*m[13] - m[8]*m[1]*m[14] + m[8]*m[2]*m[13] + m[12]*m[1]*m[10] - m[12]*m[2]*m[9];
    inv[2]  =  m[1]*m[6]*m[15]  - m[1]*m[7]*m[14]  - m[5]*m[2]*m[15] + m[5]*m[3]*m[14] + m[13]*m[2]*m[7]  - m[13]*m[3]*m[6];
    inv[6]  = -m[0]*m[6]*m[15]  + m[0]*m[7]*m[14]  + m[4]*m[2]*m[15] - m[4]*m[3]*m[14] - m[12]*m[2]*m[7]  + m[12]*m[3]*m[6];
    inv[10] =  m[0]*m[5]*m[15]  - m[0]*m[7]*m[13]  - m[4]*m[1]*m[15] + m[4]*m[3]*m[13] + m[12]*m[1]*m[7]  - m[12]*m[3]*m[5];
    inv[14] = -m[0]*m[5]*m[14]  + m[0]*m[6]*m[13]  + m[4]*m[1]*m[14] - m[4]*m[2]*m[13] - m[12]*m[1]*m[6]  + m[12]*m[2]*m[5];
    inv[3]  = -m[1]*m[6]*m[11]  + m[1]*m[7]*m[10]  + m[5]*m[2]*m[11] - m[5]*m[3]*m[10] - m[9]*m[2]*m[7]   + m[9]*m[3]*m[6];
    inv[7]  =  m[0]*m[6]*m[11]  - m[0]*m[7]*m[10]  - m[4]*m[2]*m[11] + m[4]*m[3]*m[10] + m[8]*m[2]*m[7]   - m[8]*m[3]*m[6];
    inv[11] = -m[0]*m[5]*m[11]  + m[0]*m[7]*m[9]   + m[4]*m[1]*m[11] - m[4]*m[3]*m[9]  - m[8]*m[1]*m[7]   + m[8]*m[3]*m[5];
    inv[15] =  m[0]*m[5]*m[10]  - m[0]*m[6]*m[9]   - m[4]*m[1]*m[10] + m[4]*m[2]*m[9]  + m[8]*m[1]*m[6]   - m[8]*m[2]*m[5];
    float det = m[0]*inv[0] + m[1]*inv[4] + m[2]*inv[8] + m[3]*inv[12];
    float r = 1.0f / det;
    for (int i = 0; i < 16; ++i) o[i] = inv[i] * r;
}

__global__ void pose_kernel(const float* __restrict__ pose0,
                            const float* __restrict__ pose1,
                            float* __restrict__ pose_out) {
    int b = threadIdx.x;
    if (b >= B_) return;
    float m1[16], i1[16];
    for (int i = 0; i < 16; ++i) m1[i] = pose1[b * 16 + i];
    inv4x4(m1, i1);
    for (int r = 0; r < 4; ++r)
        for (int c = 0; c < 4; ++c) {
            float s = 0.0f;
            for (int k = 0; k < 4; ++k) s += i1[r * 4 + k] * pose0[b * 16 + k * 4 + c];
            pose_out[b * 16 + r * 4 + c] = s;
        }
}

// ---------------- WMMA affine transform: 16 points per v_wmma_f32_16x16x4_f32 ----
// A (16x4): point n row = (x,y,z,1). Lane n (n<16) holds {K0,K1}={x,y};
//           lane n+16 holds {K2,K3}={z,1}  (32-bit A layout).
// B (4x16): column e = pose row e -> D[n][e] = x*R[e][0]+y*R[e][1]+z*R[e][2]+t[e].
// D (16x16 f32, 8 VGPRs): lane<16: (M=r, N=lane); lane>=16: (M=r+8, N=lane-16).
// Lanes with N<3 write their 8 consecutive points' coordinate plane (two b128s).
//
// A-fragment loads are branch-free: every lane issues two b32 loads at
// lane-computed element offsets (2*hi) and (1+hi) -- both always in-bounds --
// then a cndmask substitutes the constant 1.0 on hi lanes. The loop is
// software-pipelined so next-iteration loads overlap the current WMMA.
__global__ void transform_wmma(const float* __restrict__ src, int stride, int P,
                               const float* __restrict__ pose,
                               float* __restrict__ dx, float* __restrict__ dy,
                               float* __restrict__ dz) {
    const int lane = threadIdx.x & 31;
    const int wave = threadIdx.x >> 5;
    const int e  = lane & 15;   // N (output coord) and M (point within 16)
    const int hi = lane >> 4;
    const int pointsPerBlock = 256 * (blockDim.x >> 5);
    const int blocksPerBatch = P / pointsPerBlock;
    const int b = blockIdx.x / blocksPerBatch;
    const long pbase = (long)(blockIdx.x % blocksPerBatch) * pointsPerBlock + wave * 256;
    const float* sp = src + (long)b * P * stride;
    const float* pm = pose + b * 16;

    v2f bf;
    if (e < 4) { bf[0] = pm[e * 4 + 2 * hi + 0]; bf[1] = pm[e * 4 + 2 * hi + 1]; }
    else       { bf[0] = 0.0f; bf[1] = 0.0f; }

    const int off0 = 2 * hi;        // lo: x ; hi: z
    const int off1 = 1 + hi;        // lo: y ; hi: z (dummy, replaced by 1.0)
    const float* q = sp + (pbase + e) * stride;
    float a0 = q[off0];
    float a1 = q[off1];

#pragma unroll
    for (int it = 0; it < 16; ++it) {
        v2f af;
        af[0] = a0;
        af[1] = hi ? 1.0f : a1;
        if (it + 1 < 16) {          // issue next iteration's loads before the WMMA
            const float* qn = sp + (pbase + (it + 1) * 16 + e) * stride;
            a0 = qn[off0];
            a1 = qn[off1];
        }
        v8f c = {};
        v8f d = __builtin_amdgcn_wmma_f32_16x16x4_f32(
            false, af, false, bf, (short)0, c, false, false);
        if (e < 3) {
            float* dst = (e == 0) ? dx : (e == 1) ? dy : dz;
            long o = (long)b * P + pbase + it * 16 + hi * 8;
            v4f lo = {d[0], d[1], d[2], d[3]};
            v4f hi4 = {d[4], d[5], d[6], d[7]};
            *(v4f*)(dst + o) = lo;
            *(v4f*)(dst + o + 4) = hi4;
        }
    }
}

// ---------------- Gaussian splat: scatter-max 7x7 per radar point -----------------
// All splat values > 0 and heat initialized to 0, so uint-bit atomicMax == float max.
__global__ void heatmap_kernel(const float* __restrict__ xs, const float* __restrict__ ys,
                               const float* __restrict__ zs, int stride, int P,
                               float* __restrict__ heat) {
    long t = (long)blockIdx.x * blockDim.x + threadIdx.x;
    if (t >= (long)B_ * P) return;
    int b = (int)(t / P);
    float x = xs[t * stride], y = ys[t * stride], z = zs[t * stride];
    float px = (x - XMIN_) * INV_V;
    float py = (y - YMIN_) * INV_V;
    int ix = (int)floorf(px), iy = (int)floorf(py);
    if (ix < 0 || ix >= H_ || iy < 0 || iy >= W_ || z < ZMIN_ || z >= ZMAX_) return;
    unsigned int* hp = (unsigned int*)(heat + (long)b * H_ * W_);
#pragma unroll
    for (int dxo = -3; dxo <= 3; ++dxo) {
        int gx = ix + dxo;
        if (gx < 0 || gx >= H_) continue;
        float cx = (float)gx + 0.5f - px;
        float cx2 = cx * cx;
#pragma unroll
        for (int dyo = -3; dyo <= 3; ++dyo) {
            int gy = iy + dyo;
            if (gy < 0 || gy >= W_) continue;
            float cy = (float)gy + 0.5f - py;
            float v = __expf(-(cx2 + cy * cy) * GAUSS_K);
            atomicMax(hp + gx * W_ + gy, __float_as_uint(v));
        }
    }
}

// ---------------- Gather heatmap at point voxels ---------------------------------
__global__ void gather_kernel(const float* __restrict__ xs, const float* __restrict__ ys,
                              const float* __restrict__ zs, int stride, int P,
                              const float* __restrict__ heat,
                              float* __restrict__ out, long row_stride, long sec_off) {
    long t = (long)blockIdx.x * blockDim.x + threadIdx.x;
    if (t >= (long)B_ * P) return;
    int b = (int)(t / P);
    long i = t - (long)b * P;
    float x = xs[t * stride], y = ys[t * stride], z = zs[t * stride];
    float px = (x - XMIN_) * INV_V;
    float py = (y - YMIN_) * INV_V;
    int ix = (int)floorf(px), iy = (int)floorf(py);
    bool valid = (ix >= 0) && (ix < H_) && (iy >= 0) && (iy < W_) && (z >= ZMIN_) && (z < ZMAX_);
    int cx = min(max(ix, 0), H_ - 1);
    int cy = min(max(iy, 0), W_ - 1);
    float s = heat[(long)b * H_ * W_ + cx * W_ + cy];
    out[(long)b * row_stride + sec_off + i] = valid ? s : 0.0f;
}

extern "C" void kernel_launch(void* const* d_in, const int* in_sizes, int n_in,
                              void* d_out, int out_size, void* d_ws, size_t ws_size,
                              hipStream_t stream) {
    const float* pc0    = (const float*)d_in[0];
    const float* pc1    = (const float*)d_in[1];
    const float* radar0 = (const float*)d_in[2];
    const float* radar1 = (const float*)d_in[3];
    const float* pose0  = (const float*)d_in[4];
    const float* pose1  = (const float*)d_in[5];
    float* out = (float*)d_out;
    float* ws  = (float*)d_ws;

    // Workspace layout (floats): pose(128) | heat(B*H*W) | tpc0 x/y/z | trad0 x/y/z
    float* poseT = ws;
    float* heat  = ws + 128;
    float* tx = heat + (long)B_ * H_ * W_;
    float* ty = tx + (long)B_ * N_;
    float* tz = ty + (long)B_ * N_;
    float* rx = tz + (long)B_ * N_;
    float* ry = rx + (long)B_ * M_;
    float* rz = ry + (long)B_ * M_;
    const long row_stride = 2L * N_ + 2L * M_;

    pose_kernel<<<1, 32, 0, stream>>>(pose0, pose1, poseT);

    // WMMA transforms: 128 threads = 4 waves, 256 pts/wave -> 1024 pts/block
    transform_wmma<<<(B_ * N_) / 1024, 128, 0, stream>>>(pc0, 3, N_, poseT, tx, ty, tz);
    transform_wmma<<<(B_ * M_) / 1024, 128, 0, stream>>>(radar0, 6, M_, poseT, rx, ry, rz);

    const size_t heat_bytes = (size_t)B_ * H_ * W_ * sizeof(float);

    // Frame 0: heat from transformed radar0, gather lidar0 + radar0
    hipMemsetAsync(heat, 0, heat_bytes, stream);
    heatmap_kernel<<<(B_ * M_ + 255) / 256, 256, 0, stream>>>(rx, ry, rz, 1, M_, heat);
    gather_kernel<<<(B_ * N_ + 255) / 256, 256, 0, stream>>>(tx, ty, tz, 1, N_, heat, out, row_stride, 0);
    gather_kernel<<<(B_ * M_ + 255) / 256, 256, 0, stream>>>(rx, ry, rz, 1, M_, heat, out, row_stride, N_);

    // Frame 1: heat from raw radar1, gather lidar1 + radar1 (no transform)
    hipMemsetAsync(heat, 0, heat_bytes, stream);
    heatmap_kernel<<<(B_ * M_ + 255) / 256, 256, 0, stream>>>(radar1, radar1 + 1, radar1 + 2, 6, M_, heat);
    gather_kernel<<<(B_ * N_ + 255) / 256, 256, 0, stream>>>(pc1, pc1 + 1, pc1 + 2, 3, N_, heat, out, row_stride, (long)N_ + M_);
    gather_kernel<<<(B_ * M_ + 255) / 256, 256, 0, stream>>>(radar1, radar1 + 1, radar1 + 2, 6, M_, heat, out, row_stride, 2L * N_ + M_);
}